// MultiGraphSage_50740743635551
// MI455X (gfx1250) — compile-verified
//
#include <hip/hip_runtime.h>
#include <math.h>

typedef __attribute__((ext_vector_type(2))) float v2f;
typedef __attribute__((ext_vector_type(8))) float v8f;

#define NN 50000
#define EE 800000
#define DIN 128
#define DHID 256
#define DOUT 128

__device__ __forceinline__ void atomAddF(float* p, float v) {
    // Guaranteed hardware fp32 atomic (no CAS loop). No return value -> STOREcnt
    // only; S_ENDPGM's implicit wait-idle orders it before the next kernel.
    asm volatile("global_atomic_add_f32 %0, %1, off" : : "v"(p), "v"(v) : "memory");
}

__global__ void zero_kernel(float* __restrict__ p, int n) {
    int i = blockIdx.x * blockDim.x + threadIdx.x;
    int stride = gridDim.x * blockDim.x;
    for (; i < n; i += stride) p[i] = 0.0f;
}

__global__ void count_kernel(const int* __restrict__ dst, float* __restrict__ cnt, int E) {
    int e = blockIdx.x * blockDim.x + threadIdx.x;
    if (e < E) atomAddF(&cnt[dst[e]], 1.0f);
}

// one thread per (edge, 4-float chunk); D is a power of two so chunk = idx & mask
__global__ void scatter_kernel(const float* __restrict__ x, const int* __restrict__ src,
                               const int* __restrict__ dst, float* __restrict__ agg,
                               int E, int shift /* log2(D/4) */, int D) {
    long long idx = (long long)blockIdx.x * blockDim.x + threadIdx.x;
    int e = (int)(idx >> shift);
    if (e >= E) return;
    int c = ((int)idx & ((1 << shift) - 1)) << 2;  // float offset within row
    const float4 v = *(const float4*)(x + (size_t)src[e] * D + c);
    float* a = agg + (size_t)dst[e] * D + c;
    atomAddF(a + 0, v.x);
    atomAddF(a + 1, v.y);
    atomAddF(a + 2, v.z);
    atomAddF(a + 3, v.w);
}

// out[m,n] = tanh( (agg[m,:]/max(cnt,1)) . Wl[n,:] + x[m,:] . Wr[n,:] + bias[n] )
// grid: (NN/16, OUTD/128), block: 128 (4 waves); each wave owns a 16x32 tile.
__global__ __launch_bounds__(128) void sage_gemm_kernel(
    const float* __restrict__ agg, const float* __restrict__ cnt,
    const float* __restrict__ xin,
    const float* __restrict__ Wl, const float* __restrict__ Wr,
    const float* __restrict__ bias, float* __restrict__ out,
    int KD, int OUTD) {
    const int lane = threadIdx.x & 31;
    const int wave = threadIdx.x >> 5;
    const int lm   = lane & 15;
    const int half = lane >> 4;          // 0: K={k,k+1}, 1: K={k+2,k+3}
    const int row0 = blockIdx.x * 16;
    const int col0 = blockIdx.y * 128 + wave * 32;
    const int rowA = row0 + lm;

    const float rscale = 1.0f / fmaxf(cnt[rowA], 1.0f);

    const float* arow = agg + (size_t)rowA * KD + 2 * half;
    const float* xrow = xin + (size_t)rowA * KD + 2 * half;
    const float* wl0  = Wl + (size_t)(col0 + lm) * KD + 2 * half;
    const float* wl1  = Wl + (size_t)(col0 + 16 + lm) * KD + 2 * half;
    const float* wr0  = Wr + (size_t)(col0 + lm) * KD + 2 * half;
    const float* wr1  = Wr + (size_t)(col0 + 16 + lm) * KD + 2 * half;

    v8f c0 = {};
    v8f c1 = {};

    // phase 1: mean-aggregated features x Wl^T
    for (int k = 0; k < KD; k += 4) {
        v2f a = *(const v2f*)(arow + k);
        a = a * rscale;
        v2f b0 = *(const v2f*)(wl0 + k);
        v2f b1 = *(const v2f*)(wl1 + k);
        c0 = __builtin_amdgcn_wmma_f32_16x16x4_f32(false, a, false, b0, (short)0, c0, false, false);
        c1 = __builtin_amdgcn_wmma_f32_16x16x4_f32(false, a, false, b1, (short)0, c1, false, false);
    }
    // phase 2: self features x Wr^T (same accumulators)
    for (int k = 0; k < KD; k += 4) {
        v2f a  = *(const v2f*)(xrow + k);
        v2f b0 = *(const v2f*)(wr0 + k);
        v2f b1 = *(const v2f*)(wr1 + k);
        c0 = __builtin_amdgcn_wmma_f32_16x16x4_f32(false, a, false, b0, (short)0, c0, false, false);
        c1 = __builtin_amdgcn_wmma_f32_16x16x4_f32(false, a, false, b1, (short)0, c1, false, false);
    }

    const float bn0 = bias[col0 + lm];
    const float bn1 = bias[col0 + 16 + lm];
#pragma unroll
    for (int r = 0; r < 8; ++r) {
        int m = row0 + r + 8 * half;   // documented 16x16 f32 C/D layout
        out[(size_t)m * OUTD + col0 + lm]      = tanhf(c0[r] + bn0);
        out[(size_t)m * OUTD + col0 + 16 + lm] = tanhf(c1[r] + bn1);
    }
}

extern "C" void kernel_launch(void* const* d_in, const int* in_sizes, int n_in,
                              void* d_out, int out_size, void* d_ws, size_t ws_size,
                              hipStream_t stream) {
    (void)in_sizes; (void)n_in; (void)out_size; (void)ws_size;

    // workspace layout (floats): agg[N*DHID] | cnt[N] | h[N*DHID]
    float* agg = (float*)d_ws;
    float* cnt = agg + (size_t)NN * DHID;
    float* h   = cnt + NN;
    float* out = (float*)d_out;

    for (int g = 0; g < 2; ++g) {
        const float* x  = (const float*)d_in[g];
        const int*  src = (const int*)d_in[2 + g];
        const int*  dst = src + EE;
        int base = 4 + 6 * g;
        const float* W1l = (const float*)d_in[base + 0];
        const float* b1  = (const float*)d_in[base + 1];
        const float* W1r = (const float*)d_in[base + 2];
        const float* W2l = (const float*)d_in[base + 3];
        const float* b2  = (const float*)d_in[base + 4];
        const float* W2r = (const float*)d_in[base + 5];
        float* outg = out + (size_t)g * NN * DOUT;

        // ---- layer 1 ----
        zero_kernel<<<2048, 256, 0, stream>>>(agg, NN * DIN);
        zero_kernel<<<256, 256, 0, stream>>>(cnt, NN);
        count_kernel<<<(EE + 255) / 256, 256, 0, stream>>>(dst, cnt, EE);
        {
            long long total = (long long)EE * (DIN / 4);
            scatter_kernel<<<(unsigned)((total + 255) / 256), 256, 0, stream>>>(
                x, src, dst, agg, EE, 5 /* log2(128/4) */, DIN);
        }
        sage_gemm_kernel<<<dim3(NN / 16, DHID / 128), 128, 0, stream>>>(
            agg, cnt, x, W1l, W1r, b1, h, DIN, DHID);

        // ---- layer 2 (counts unchanged: same dst list) ----
        zero_kernel<<<2048, 256, 0, stream>>>(agg, NN * DHID);
        {
            long long total = (long long)EE * (DHID / 4);
            scatter_kernel<<<(unsigned)((total + 255) / 256), 256, 0, stream>>>(
                h, src, dst, agg, EE, 6 /* log2(256/4) */, DHID);
        }
        sage_gemm_kernel<<<dim3(NN / 16, DOUT / 128), 128, 0, stream>>>(
            agg, cnt, h, W2l, W2r, b2, outg, DHID, DOUT);
    }
}